// MultiLayerBase_29326036697852
// MI455X (gfx1250) — compile-verified
//
#include <hip/hip_runtime.h>
#include <math.h>

typedef __attribute__((ext_vector_type(2))) float v2f;
typedef __attribute__((ext_vector_type(8))) float v8f;
typedef unsigned int u32x4 __attribute__((ext_vector_type(4)));
typedef int i32x4 __attribute__((ext_vector_type(4)));
typedef int i32x8 __attribute__((ext_vector_type(8)));

#define Bsz 64
#define Ssz 1024
#define Hsz 256
#define G3  768
#define LDP 260   // padded LDS row pitch in floats (260 % 64 == 4 -> conflict-free A reads)

// ---- TDM availability probing (toolchain-dependent) -------------------------
#if defined(__has_builtin)
#  if __has_builtin(__builtin_amdgcn_tensor_load_to_lds)
#    define HAVE_TDM 1
#  endif
#endif
#ifndef HAVE_TDM
#  define HAVE_TDM 0
#endif
#if __has_include(<hip/amd_detail/amd_gfx1250_TDM.h>)
#  define TDM_6ARG 1
#else
#  define TDM_6ARG 0
#endif

#if HAVE_TDM
#  pragma message("CDNA5 probe: __builtin_amdgcn_tensor_load_to_lds AVAILABLE in this pass")
#else
#  pragma message("CDNA5 probe: TDM builtin NOT available in this pass (float4 staging fallback)")
#endif
#if TDM_6ARG
#  pragma message("CDNA5 probe: using 6-arg TDM form (therock headers present)")
#else
#  pragma message("CDNA5 probe: using 5-arg TDM form (ROCm 7.2 style)")
#endif

__device__ __forceinline__ float fsig(float x) { return 1.0f / (1.0f + __expf(-x)); }
__device__ __forceinline__ float ftanh(float x) { return 2.0f * fsig(2.0f * x) - 1.0f; }

__device__ __forceinline__ v8f wmma_f32(v2f a, v2f b, v8f c) {
  // D = A(16x4 f32) * B(4x16 f32) + C(16x16 f32)
  return __builtin_amdgcn_wmma_f32_16x16x4_f32(false, a, false, b, (short)0, c, false, false);
}

#if HAVE_TDM
// 2-D TDM tile load: tile_h rows x tile_w f32 elements, global row stride
// `stride_elems`, LDS destination padded +4 DWORDs every 256 DWORDs (pitch 260
// for 256-wide tiles). D# layout per CDNA5 ISA ch.8.
__device__ __forceinline__ void tdm_load_2d(unsigned lds_off, const void* gptr,
                                            unsigned tile_w, unsigned tile_h,
                                            unsigned tensor_w,
                                            unsigned long long stride_elems) {
  unsigned long long ga = (unsigned long long)gptr;
  u32x4 g0 = { 1u,                                   // count=1, user descriptor
               lds_off,                              // lds_addr [63:32]
               (unsigned)(ga & 0xffffffffu),         // global_addr[31:0]
               (unsigned)((ga >> 32) & 0x01ffffffu) | (2u << 30) }; // addr[56:32] | type=2
  const unsigned DIM1 = 1u << 20;                    // tensor_dim1 (rows; OOB bound only)
  unsigned w0 = (2u << 16)                           // data_size = 4 bytes
              | (1u << 20)                           // pad_enable
              | (7u << 22)                           // pad_interval: every 256 DWORDs
              | (3u << 25);                          // pad_amount: 4 DWORDs
  unsigned w1 = (tensor_w & 0xffffu) << 16;                          // tensor_dim0[15:0]
  unsigned w2 = ((tensor_w >> 16) & 0xffffu) | ((DIM1 & 0xffffu) << 16);
  unsigned w3 = ((DIM1 >> 16) & 0xffffu) | ((tile_w & 0xffffu) << 16);
  unsigned w4 = (tile_h & 0xffffu);                                  // tile_dim1; tile_dim2=0
  unsigned w5 = (unsigned)(stride_elems & 0xffffffffull);            // dim0_stride[31:0]
  unsigned w6 = (unsigned)((stride_elems >> 32) & 0xffffull);        // dim0_stride[47:32]
  i32x8 g1 = { (int)w0, (int)w1, (int)w2, (int)w3, (int)w4, (int)w5, (int)w6, 0 };
  i32x4 z4 = { 0, 0, 0, 0 };
#if TDM_6ARG
  i32x8 z8 = { 0, 0, 0, 0, 0, 0, 0, 0 };
  __builtin_amdgcn_tensor_load_to_lds(g0, g1, z4, z4, z8, 0);
#else
  __builtin_amdgcn_tensor_load_to_lds(g0, g1, z4, z4, 0);
#endif
}
#endif // HAVE_TDM

// ---------------------------------------------------------------------------
// Kernel 1: whole-GPU parallel input projection  xg = x @ W_ih^T  (no bias).
// Grid (3, BS/16): blockIdx.x = 256-col chunk of 3H, blockIdx.y = 16-row tile.
// Block = 512 threads (16 waves); wave w owns output cols [x*256+16w, +16).
// A tile staged by the Tensor Data Mover with LDS padding -> pitch 260.
// ---------------------------------------------------------------------------
__global__ __launch_bounds__(512)
void xg_gemm_kernel(const float* __restrict__ in, int ip,
                    const float* __restrict__ Wih, float* __restrict__ xg) {
  __shared__ float lds_a[16][LDP];

  const int tid   = threadIdx.x;
  const int wv    = tid >> 5;
  const int ln    = tid & 31;
  const int n     = ln & 15;
  const int koff  = (ln < 16) ? 0 : 2;
  const int m     = n;
  const int mrow0 = (ln < 16) ? 0 : 8;
  const int r0    = blockIdx.y * 16;                     // flattened (b*S+t) row tile
  const int col   = blockIdx.x * 256 + wv * 16 + n;      // output column 0..767

  const float* pW = Wih + (size_t)col * Hsz + koff;      // B[k][n] = Wih[col][k]

#if HAVE_TDM
  if (wv == 0) {
    tdm_load_2d((unsigned)(unsigned long long)&lds_a[0][0],
                in + (size_t)r0 * ip, /*tile_w=*/256, /*tile_h=*/16,
                /*tensor_w=*/(unsigned)ip, /*stride=*/(unsigned long long)ip);
    __builtin_amdgcn_s_wait_tensorcnt(0);
  }
#else
  for (int j = tid; j < 16 * 64; j += 512) {
    int rr = j >> 6, c4 = (j & 63) << 2;
    *(float4*)&lds_a[rr][c4] = *(const float4*)(in + (size_t)(r0 + rr) * ip + c4);
  }
#endif
  __syncthreads();

  v8f acc = {};
#pragma unroll 4
  for (int k0 = 0; k0 < Hsz; k0 += 4) {
    v2f a = *(const v2f*)&lds_a[m][k0 + koff];
    v2f b = *(const v2f*)(pW + k0);
    acc = wmma_f32(a, b, acc);
  }

#pragma unroll
  for (int v = 0; v < 8; ++v)
    xg[(size_t)(r0 + mrow0 + v) * G3 + col] = acc[v];
}

// ---------------------------------------------------------------------------
// Kernel 2: recurrent scan using precomputed xg. Hidden-side GEMM only:
// 192 v_wmma_f32_16x16x4_f32 per wave per timestep. 4 blocks (batch tiles),
// 16 waves each. Double-buffered LDS h state -> ONE barrier per timestep.
// ---------------------------------------------------------------------------
__global__ __launch_bounds__(512)
void gru_scan_kernel(const float* __restrict__ xg,       // [B*S, 768]
                     const float* __restrict__ Whh,      // [3H, 256]
                     const float* __restrict__ bih,
                     const float* __restrict__ bhh,
                     float* __restrict__ outA, int pa,
                     float* __restrict__ outB, int pb) {
  __shared__ float lds_h[2][16][LDP];

  const int tid   = threadIdx.x;
  const int wv    = tid >> 5;
  const int ln    = tid & 31;
  const int n     = ln & 15;
  const int koff  = (ln < 16) ? 0 : 2;
  const int m     = n;
  const int mrow0 = (ln < 16) ? 0 : 8;
  const int b0    = blockIdx.x * 16;
  const int c     = wv * 16 + n;

  for (int i = tid; i < 16 * LDP; i += 512) ((float*)lds_h[0])[i] = 0.0f;

  const float* pHr = Whh + (size_t)(c)           * Hsz + koff;
  const float* pHz = Whh + (size_t)(Hsz + c)     * Hsz + koff;
  const float* pHn = Whh + (size_t)(2 * Hsz + c) * Hsz + koff;

  const float bR  = bih[c] + bhh[c];
  const float bZ  = bih[Hsz + c] + bhh[Hsz + c];
  const float bNi = bih[2 * Hsz + c];
  const float bNh = bhh[2 * Hsz + c];

  __syncthreads();

  int cur = 0;
  for (int t = 0; t < Ssz; ++t) {
    // Issue precomputed input-gate loads early; consumed after ~192 WMMAs of
    // latency hiding (xg is L2-resident: 192MB L2 holds the whole tensor).
    float xr[8], xz[8], xn[8];
#pragma unroll
    for (int v = 0; v < 8; ++v) {
      const float* p = xg + ((size_t)(b0 + mrow0 + v) * Ssz + t) * G3;
      xr[v] = p[c];
      xz[v] = p[Hsz + c];
      xn[v] = p[2 * Hsz + c];
    }

    v8f accR = {}, accZ = {}, accN = {};
#pragma unroll 4
    for (int k0 = 0; k0 < Hsz; k0 += 4) {
      v2f a  = *(const v2f*)&lds_h[cur][m][k0 + koff];
      v2f br = *(const v2f*)(pHr + k0);
      v2f bz = *(const v2f*)(pHz + k0);
      v2f bn = *(const v2f*)(pHn + k0);
      accR = wmma_f32(a, br, accR);
      accZ = wmma_f32(a, bz, accZ);
      accN = wmma_f32(a, bn, accN);
    }

    // Gates + state update; writes go to the OTHER h buffer, so no
    // read-phase barrier is needed — just one end-of-step barrier.
#pragma unroll
    for (int v = 0; v < 8; ++v) {
      const int row = mrow0 + v;
      const float hp = lds_h[cur][row][c];
      const float r  = fsig(accR[v] + xr[v] + bR);
      const float z  = fsig(accZ[v] + xz[v] + bZ);
      const float nn = ftanh(xn[v] + bNi + r * (accN[v] + bNh));
      const float hN = (1.0f - z) * nn + z * hp;
      lds_h[cur ^ 1][row][c] = hN;
      const size_t o = (size_t)(b0 + row) * Ssz + t;
      outA[o * (size_t)pa + c] = hN;
      if (outB) outB[o * (size_t)pb + c] = hN;
    }
    __syncthreads();   // h_t visible before next step's GEMM
    cur ^= 1;
  }
}

// ---------------------------------------------------------------------------
// Fallback: fused single-kernel layer (input GEMM + scan), used when d_ws is
// too small for the 192MB xg scratch.
// ---------------------------------------------------------------------------
__global__ __launch_bounds__(512)
void gru_layer_fused_kernel(const float* __restrict__ in, int ip,
                            const float* __restrict__ Wih,
                            const float* __restrict__ Whh,
                            const float* __restrict__ bih,
                            const float* __restrict__ bhh,
                            float* __restrict__ outA, int pa,
                            float* __restrict__ outB, int pb) {
  __shared__ float lds_x[16][LDP];
  __shared__ float lds_h[2][16][LDP];

  const int tid   = threadIdx.x;
  const int wv    = tid >> 5;
  const int ln    = tid & 31;
  const int n     = ln & 15;
  const int koff  = (ln < 16) ? 0 : 2;
  const int m     = n;
  const int mrow0 = (ln < 16) ? 0 : 8;
  const int b0    = blockIdx.x * 16;
  const int c     = wv * 16 + n;

  for (int i = tid; i < 16 * LDP; i += 512) ((float*)lds_h[0])[i] = 0.0f;

  const float* pIr = Wih + (size_t)(c)           * Hsz + koff;
  const float* pIz = Wih + (size_t)(Hsz + c)     * Hsz + koff;
  const float* pIn = Wih + (size_t)(2 * Hsz + c) * Hsz + koff;
  const float* pHr = Whh + (size_t)(c)           * Hsz + koff;
  const float* pHz = Whh + (size_t)(Hsz + c)     * Hsz + koff;
  const float* pHn = Whh + (size_t)(2 * Hsz + c) * Hsz + koff;

  const float bR  = bih[c] + bhh[c];
  const float bZ  = bih[Hsz + c] + bhh[Hsz + c];
  const float bNi = bih[2 * Hsz + c];
  const float bNh = bhh[2 * Hsz + c];

  int cur = 0;
  for (int t = 0; t < Ssz; ++t) {
    for (int j = tid; j < 16 * 64; j += 512) {
      int rr = j >> 6, c4 = (j & 63) << 2;
      const float4 v = *(const float4*)(in + ((size_t)(b0 + rr) * Ssz + t) * ip + c4);
      *(float4*)&lds_x[rr][c4] = v;
    }
    if (tid < 16 && t + 1 < Ssz)
      __builtin_prefetch(in + ((size_t)(b0 + tid) * Ssz + (t + 1)) * ip, 0, 0);
    __syncthreads();

    v8f accR = {}, accZ = {}, accNx = {}, accNh = {};
#pragma unroll 4
    for (int k0 = 0; k0 < Hsz; k0 += 4) {
      v2f a  = *(const v2f*)&lds_x[m][k0 + koff];
      v2f br = *(const v2f*)(pIr + k0);
      v2f bz = *(const v2f*)(pIz + k0);
      v2f bn = *(const v2f*)(pIn + k0);
      accR  = wmma_f32(a, br, accR);
      accZ  = wmma_f32(a, bz, accZ);
      accNx = wmma_f32(a, bn, accNx);
    }
#pragma unroll 4
    for (int k0 = 0; k0 < Hsz; k0 += 4) {
      v2f a  = *(const v2f*)&lds_h[cur][m][k0 + koff];
      v2f br = *(const v2f*)(pHr + k0);
      v2f bz = *(const v2f*)(pHz + k0);
      v2f bn = *(const v2f*)(pHn + k0);
      accR  = wmma_f32(a, br, accR);
      accZ  = wmma_f32(a, bz, accZ);
      accNh = wmma_f32(a, bn, accNh);
    }

#pragma unroll
    for (int v = 0; v < 8; ++v) {
      const int row = mrow0 + v;
      const float hp = lds_h[cur][row][c];
      const float r  = fsig(accR[v] + bR);
      const float z  = fsig(accZ[v] + bZ);
      const float nn = ftanh(accNx[v] + bNi + r * (accNh[v] + bNh));
      const float hN = (1.0f - z) * nn + z * hp;
      lds_h[cur ^ 1][row][c] = hN;
      const size_t o = (size_t)(b0 + row) * Ssz + t;
      outA[o * (size_t)pa + c] = hN;
      if (outB) outB[o * (size_t)pb + c] = hN;
    }
    __syncthreads();
    cur ^= 1;
  }
}

extern "C" void kernel_launch(void* const* d_in, const int* in_sizes, int n_in,
                              void* d_out, int out_size, void* d_ws, size_t ws_size,
                              hipStream_t stream) {
  (void)in_sizes; (void)n_in; (void)out_size;

  const float* x    = (const float*)d_in[0];
  const float* Wih0 = (const float*)d_in[1];
  const float* Whh0 = (const float*)d_in[2];
  const float* bih0 = (const float*)d_in[3];
  const float* bhh0 = (const float*)d_in[4];
  const float* Wih1 = (const float*)d_in[5];
  const float* Whh1 = (const float*)d_in[6];
  const float* bih1 = (const float*)d_in[7];
  const float* bhh1 = (const float*)d_in[8];

  float* out = (float*)d_out;
  float* h2  = out;                                        // [B,S,256]
  float* cc  = out + (size_t)Bsz * Ssz * Hsz;              // concat [B,S,512] = [h1 | h2]

  const size_t XG_BYTES = (size_t)Bsz * Ssz * G3 * sizeof(float);  // 192 MB

  if (ws_size >= XG_BYTES) {
    float* xg = (float*)d_ws;
    const dim3 gG(3, (Bsz * Ssz) / 16);    // 12288 blocks: whole-GPU GEMM
    // Layer 0
    xg_gemm_kernel<<<gG, dim3(512), 0, stream>>>(x, 256, Wih0, xg);
    gru_scan_kernel<<<dim3(4), dim3(512), 0, stream>>>(
        xg, Whh0, bih0, bhh0, cc, 512, (float*)nullptr, 0);
    // Layer 1 (input = h1 living in concat's first half, pitch 512)
    xg_gemm_kernel<<<gG, dim3(512), 0, stream>>>(cc, 512, Wih1, xg);
    gru_scan_kernel<<<dim3(4), dim3(512), 0, stream>>>(
        xg, Whh1, bih1, bhh1, h2, 256, cc + 256, 512);
  } else {
    gru_layer_fused_kernel<<<dim3(4), dim3(512), 0, stream>>>(
        x, 256, Wih0, Whh0, bih0, bhh0, cc, 512, (float*)nullptr, 0);
    gru_layer_fused_kernel<<<dim3(4), dim3(512), 0, stream>>>(
        cc, 512, Wih1, Whh1, bih1, bhh1, h2, 256, cc + 256, 512);
  }
}